// EdgeModel_67791763800205
// MI455X (gfx1250) — compile-verified
//
#include <hip/hip_runtime.h>

typedef __attribute__((ext_vector_type(2))) float v2f;
typedef __attribute__((ext_vector_type(8))) float v8f;

#define TILE_E   64
#define KC       64          // hidden-dim chunk
#define HDIM     512
#define OUTF     512
#define XCOLS    36          // 35 features + 1 zero pad
#define XSTRIDE  38          // even -> 8B-aligned k-pairs, 16 distinct banks
#define HSTRIDE  66
#define NODE_F   9
#define U_F      16
#define W1P_ROWS 18          // ceil(36/2) k-pairs (row 35 zero-padded)
#define W2P_ROWS 256         // 512/2 k-pairs

// ---------------- pre-pass: pack W1/W2 into k-pair-interleaved layout --------
// w1p[k2*512+n] = (W1[2k2][n], W1[2k2+1][n])  (rows >=35 are zero)
// w2p[k2*512+n] = (W2[2k2][n], W2[2k2+1][n])
__global__ __launch_bounds__(256)
void pack_pairs_kernel(const float* __restrict__ W1,
                       const float* __restrict__ W2,
                       v2f* __restrict__ w1p,
                       v2f* __restrict__ w2p)
{
    int idx = blockIdx.x * 256 + threadIdx.x;
    int n1 = W1P_ROWS * HDIM;            // 9216
    int n2 = W2P_ROWS * OUTF;            // 131072
    if (idx < n1) {
        int k2 = idx / HDIM, n = idx - k2 * HDIM;
        int k0 = 2 * k2, k1 = 2 * k2 + 1;
        v2f p;
        p.x = (k0 < 35) ? W1[k0 * HDIM + n] : 0.0f;
        p.y = (k1 < 35) ? W1[k1 * HDIM + n] : 0.0f;
        w1p[idx] = p;
    } else if (idx < n1 + n2) {
        int j = idx - n1;
        int k2 = j / OUTF, n = j - k2 * OUTF;
        v2f p;
        p.x = W2[(2 * k2)     * OUTF + n];
        p.y = W2[(2 * k2 + 1) * OUTF + n];
        w2p[j] = p;
    }
}

// ---------------- fused edge MLP -------------------------------------------
template <bool PAIRED>
__global__ __launch_bounds__(256)
void edge_mlp_kernel(const float* __restrict__ src,
                     const float* __restrict__ dst,
                     const float* __restrict__ eattr,
                     const float* __restrict__ u,
                     const int*   __restrict__ batch,
                     const float* __restrict__ W1,
                     const float* __restrict__ b1,
                     const float* __restrict__ W2,
                     const float* __restrict__ b2,
                     const v2f*   __restrict__ w1p,
                     const v2f*   __restrict__ w2p,
                     float* __restrict__ out,
                     int E)
{
    __shared__ float XS[TILE_E * XSTRIDE];
    __shared__ float HS[TILE_E * HSTRIDE];

    const int tid   = threadIdx.x;
    const int wave  = tid >> 5;
    const int lane  = tid & 31;
    const int l15   = lane & 15;
    const int lhi   = lane >> 4;              // 0 or 1 (lane half)
    const int ebase = blockIdx.x * TILE_E;

    // ---- build X tile [64 x 36] in LDS (gather + concat) ----
    for (int i = 0; i < (TILE_E * XCOLS) / 256; ++i) {
        int idx = tid + i * 256;              // 2304 elements, 9 per thread
        int er  = idx / XCOLS;
        int c   = idx - er * XCOLS;
        int e   = ebase + er; if (e >= E) e = E - 1;   // clamp (guarded at store)
        float v;
        if      (c < NODE_F)      v = src[e * NODE_F + c];
        else if (c < 2 * NODE_F)  v = dst[e * NODE_F + (c - NODE_F)];
        else if (c == 18)         v = eattr[e];
        else if (c < 35)          v = u[batch[e] * U_F + (c - 19)];
        else                      v = 0.0f;   // pad column 35
        XS[er * XSTRIDE + c] = v;
    }
    __syncthreads();

    // layer-2 ownership: wave -> (M-tile of 16 edges, half of the 512 out cols)
    const int mt2    = wave >> 1;             // 0..3
    const int nbase2 = (wave & 1) * 256;      // 0 or 256

    // accumulators pre-loaded with b2 (bias folds into WMMA C operand)
    v8f acc[16];
    for (int nt = 0; nt < 16; ++nt) {
        float bv = b2[nbase2 + nt * 16 + l15];
        v8f t;
        #pragma unroll
        for (int v = 0; v < 8; ++v) t[v] = bv;
        acc[nt] = t;
    }

    const int t0 = wave * 2;                  // layer-1: 2 of 16 tiles per wave

    for (int kc = 0; kc < HDIM; kc += KC) {
        // ---- layer 1: H[:, kc:kc+64] = relu(X@W1 + b1) -> LDS ----
        #pragma unroll
        for (int i = 0; i < 2; ++i) {
            int t    = t0 + i;
            int mt   = t >> 2;                // 0..3  (16-edge row tile)
            int ntl  = t & 3;                 // 0..3  (16-col tile within chunk)
            int colg = kc + ntl * 16 + l15;   // global hidden column
            int arow = mt * 16 + l15;
            v8f hacc = {};
            #pragma unroll
            for (int ks = 0; ks < 9; ++ks) {  // K = 36 (pad), 4 per WMMA
                int ka = ks * 4 + 2 * lhi;    // even -> aligned pair
                v2f a = *(const v2f*)(XS + arow * XSTRIDE + ka);
                v2f b;
                if (PAIRED) {
                    b = w1p[(2 * ks + lhi) * HDIM + colg];   // one b64 load
                } else {
                    int kb0 = ka     > 34 ? 34 : ka;
                    int kb1 = ka + 1 > 34 ? 34 : ka + 1;
                    b.x = W1[kb0 * HDIM + colg];
                    b.y = W1[kb1 * HDIM + colg];
                }
                hacc = __builtin_amdgcn_wmma_f32_16x16x4_f32(
                           false, a, false, b, (short)0, hacc, false, false);
            }
            float bias = b1[colg];
            int   coll = ntl * 16 + l15;
            #pragma unroll
            for (int v = 0; v < 8; ++v) {
                float hv = hacc[v] + bias;
                hv = hv > 0.0f ? hv : 0.0f;   // relu
                int row = mt * 16 + v + 8 * lhi;
                HS[row * HSTRIDE + coll] = hv;
            }
        }
        __syncthreads();

        // ---- layer 2: acc += H_chunk @ W2[kc:kc+64, :] ----
        int arow2 = mt2 * 16 + l15;
        #pragma unroll
        for (int ks = 0; ks < KC / 4; ++ks) {
            int kl = ks * 4 + 2 * lhi;        // even -> aligned pair
            v2f a = *(const v2f*)(HS + arow2 * HSTRIDE + kl);
            #pragma unroll
            for (int nt = 0; nt < 16; ++nt) {
                v2f b;
                if (PAIRED) {
                    // pair row = (kc + 4ks + 2lhi)/2 = kc/2 + 2ks + lhi
                    b = w2p[((kc >> 1) + 2 * ks + lhi) * OUTF
                            + nbase2 + nt * 16 + l15];       // one b64 load
                } else {
                    const float* p = W2 + (kc + kl) * OUTF + nbase2 + l15;
                    b.x = p[nt * 16];
                    b.y = p[nt * 16 + OUTF];
                }
                acc[nt] = __builtin_amdgcn_wmma_f32_16x16x4_f32(
                              false, a, false, b, (short)0, acc[nt], false, false);
            }
        }
        __syncthreads();                      // HS reused next chunk
    }

    // ---- store 16x256 per wave (guard tail edges) ----
    for (int nt = 0; nt < 16; ++nt) {
        int col = nbase2 + nt * 16 + l15;
        #pragma unroll
        for (int v = 0; v < 8; ++v) {
            int e = ebase + mt2 * 16 + v + 8 * lhi;
            if (e < E) out[(size_t)e * OUTF + col] = acc[nt][v];
        }
    }
}

extern "C" void kernel_launch(void* const* d_in, const int* in_sizes, int n_in,
                              void* d_out, int out_size, void* d_ws, size_t ws_size,
                              hipStream_t stream) {
    const float* src   = (const float*)d_in[0];
    const float* dst   = (const float*)d_in[1];
    const float* eattr = (const float*)d_in[2];
    const float* u     = (const float*)d_in[3];
    const int*   batch = (const int*)  d_in[4];
    const float* W1    = (const float*)d_in[5];
    const float* b1    = (const float*)d_in[6];
    const float* W2    = (const float*)d_in[7];
    const float* b2    = (const float*)d_in[8];
    float* out = (float*)d_out;

    int E = in_sizes[0] / NODE_F;
    if (E <= 0) return;
    int grid = (E + TILE_E - 1) / TILE_E;

    const size_t n1 = (size_t)W1P_ROWS * HDIM;     // 9216 pairs
    const size_t n2 = (size_t)W2P_ROWS * OUTF;     // 131072 pairs
    const size_t need = (n1 + n2) * sizeof(v2f);   // ~1.1 MB

    if (ws_size >= need) {
        v2f* w1p = (v2f*)d_ws;
        v2f* w2p = w1p + n1;
        int total = (int)(n1 + n2);
        pack_pairs_kernel<<<(total + 255) / 256, 256, 0, stream>>>(W1, W2, w1p, w2p);
        edge_mlp_kernel<true><<<grid, 256, 0, stream>>>(
            src, dst, eattr, u, batch, W1, b1, W2, b2, w1p, w2p, out, E);
    } else {
        edge_mlp_kernel<false><<<grid, 256, 0, stream>>>(
            src, dst, eattr, u, batch, W1, b1, W2, b2, nullptr, nullptr, out, E);
    }
}